// PositionalEmbedding_58652073394524
// MI455X (gfx1250) — compile-verified
//
#include <hip/hip_runtime.h>

// PositionalEmbedding: out[b,n,d] = (cos(s*w_d), sin(s*w_d)) * exp(-50*min(s,0)^2)
// Output layout (complex64 interleaved): flat float index ((b*N+n)*D + d)*2.
// Per point: 12 floats = 3 float4s. One thread per float4 -> wave32 issues
// fully coalesced 512B global_store_b128 NT streams (write-once, skip L2 rinse).

typedef float v4f __attribute__((ext_vector_type(4)));

__global__ __launch_bounds__(256) void
PositionalEmbedding_58652073394524_kernel(const float* __restrict__ sdf,
                                          const float* __restrict__ w,
                                          float* __restrict__ out,
                                          int n_points /* B*N */) {
    const int total_vec4 = n_points * 3;           // 12M float4s
    int t = blockIdx.x * blockDim.x + threadIdx.x; // one float4 per thread
    if (t >= total_vec4) return;

    // point index and which third of the 12-float record we produce
    unsigned ut = (unsigned)t;
    unsigned p  = ut / 3u;          // magic-multiply, cheap
    unsigned q  = ut - 3u * p;      // 0,1,2 -> frequency pairs (0,1),(2,3),(4,5)

    float s = sdf[p];               // L2-resident re-read (3x), negligible vs stores

    // envelope: exp(-min(s,0)^2 / 0.02) = exp2(-50*log2(e) * m^2)
    float m   = fminf(s, 0.0f);
    float env = __builtin_amdgcn_exp2f(m * m * -72.13475204444817f); // v_exp_f32

    // two frequencies for this float4
    int d0   = 2 * (int)q;
    float a0 = s * w[d0];
    float a1 = s * w[d0 + 1];

    // hardware v_sin/v_cos take input in revolutions
    const float INV_2PI = 0.15915494309189535f;
    float r0 = a0 * INV_2PI;
    float r1 = a1 * INV_2PI;
    float c0 = __builtin_amdgcn_cosf(r0);   // v_cos_f32
    float s0 = __builtin_amdgcn_sinf(r0);   // v_sin_f32
    float c1 = __builtin_amdgcn_cosf(r1);
    float s1 = __builtin_amdgcn_sinf(r1);

    v4f v;
    v.x = c0 * env;   // re(d0)
    v.y = s0 * env;   // im(d0)
    v.z = c1 * env;   // re(d0+1)
    v.w = s1 * env;   // im(d0+1)

    // streaming non-temporal 128-bit store: global_store_b128 th:NT
    __builtin_nontemporal_store(v, (v4f*)out + t);
}

extern "C" void kernel_launch(void* const* d_in, const int* in_sizes, int n_in,
                              void* d_out, int out_size, void* d_ws, size_t ws_size,
                              hipStream_t stream) {
    (void)n_in; (void)d_ws; (void)ws_size; (void)out_size;

    const float* sdf = (const float*)d_in[0];  // [B,N,1] float32, B*N elements
    const float* w   = (const float*)d_in[1];  // [D] float32, D=6
    float* out       = (float*)d_out;          // [B,N,D] complex64 -> 12 floats/point

    const int n_points   = in_sizes[0];        // B*N = 4,000,000
    const int total_vec4 = n_points * 3;       // float4s to write

    const int block = 256;
    const int grid  = (total_vec4 + block - 1) / block;

    PositionalEmbedding_58652073394524_kernel<<<grid, block, 0, stream>>>(
        sdf, w, out, n_points);
}